// SlowCustomActivation_88441966559493
// MI455X (gfx1250) — compile-verified
//
#include <hip/hip_runtime.h>
#include <hip/hip_bf16.h>

// out = x>0 ? 0.9*x + 0.1*tanh(x) : 0.5*tanh(x), fp32, 8192*8192 elements.
// HBM-bandwidth-bound: 512 MB traffic => ~22us floor at 23.3 TB/s.
// b128 non-temporal load/store, hardware v_tanh_f32, 32-bit addressing
// (byte extent 256 MB < 4 GiB so u32 voffset + SGPR base suffices).

typedef __attribute__((ext_vector_type(4))) float v4f;

__device__ __forceinline__ float fast_tanh(float x) {
#if __has_builtin(__builtin_amdgcn_tanhf)
    return __builtin_amdgcn_tanhf(x);          // v_tanh_f32 (confirmed in disasm)
#elif __has_builtin(__builtin_amdgcn_tanh_f32)
    return __builtin_amdgcn_tanh_f32(x);
#else
    float e = __builtin_amdgcn_exp2f(x * 2.885390081777926f); // 2*log2(e)
    return 1.0f - 2.0f * __builtin_amdgcn_rcpf(e + 1.0f);
#endif
}

__device__ __forceinline__ float act(float x) {
    float t = fast_tanh(x);
    float pos = __builtin_fmaf(x, 0.9f, 0.1f * t);
    float neg = 0.5f * t;
    return x > 0.0f ? pos : neg;
}

// Exact-cover: one v4f per thread, u32 index (n4 <= 2^28 for this problem).
__global__ void __launch_bounds__(256)
act_vec4_kernel(const float* __restrict__ x, float* __restrict__ out, unsigned n4) {
    const v4f* __restrict__ xv = (const v4f*)x;
    v4f* __restrict__ ov = (v4f*)out;

    unsigned i = blockIdx.x * 256u + threadIdx.x;
    if (i < n4) {
        v4f v = __builtin_nontemporal_load(&xv[i]);   // global_load_b128 th:NT
        v4f r;
        r.x = act(v.x);
        r.y = act(v.y);
        r.z = act(v.z);
        r.w = act(v.w);
        __builtin_nontemporal_store(r, &ov[i]);       // global_store_b128 th:NT
    }
}

// Scalar tail for n % 4 != 0 (no-op for 8192^2); also generic fallback for
// huge n that would overflow u32 vec4 indexing.
__global__ void act_tail_kernel(const float* __restrict__ x, float* __restrict__ out,
                                long long start, long long n) {
    long long i = start + (long long)blockIdx.x * blockDim.x + threadIdx.x;
    if (i < n) {
        out[i] = act(x[i]);
    }
}

extern "C" void kernel_launch(void* const* d_in, const int* in_sizes, int n_in,
                              void* d_out, int out_size, void* d_ws, size_t ws_size,
                              hipStream_t stream) {
    const float* x = (const float*)d_in[0];
    float* out = (float*)d_out;

    long long n  = (long long)in_sizes[0];   // 67,108,864 for 8192x8192
    long long n4 = n >> 2;

    // u32 vec4 indexing is valid while n4 fits in 32 bits (true up to 16G elements).
    if (n4 > 0 && n4 <= 0xFFFFFFFFLL) {
        const int block = 256;               // 8 wave32 per block
        unsigned grid = (unsigned)((n4 + block - 1) / block);
        act_vec4_kernel<<<grid, block, 0, stream>>>(x, out, (unsigned)n4);
        long long tail_start = n4 << 2;
        long long tail = n - tail_start;
        if (tail > 0) {
            act_tail_kernel<<<1, (int)tail, 0, stream>>>(x, out, tail_start, n);
        }
    } else if (n > 0) {
        // Generic scalar path (never taken for this problem size).
        const int block = 256;
        long long blocks = (n + block - 1) / block;
        int grid = (blocks > 0x7FFFFFFFLL) ? 0x7FFFFFFF : (int)blocks;
        act_tail_kernel<<<grid, block, 0, stream>>>(x, out, 0, n);
    }
}